// EntmaxBisect_function_88055419502699
// MI455X (gfx1250) — compile-verified
//
#include <hip/hip_runtime.h>
#include <math.h>

// Entmax-bisect, one workgroup (256 threads, 8 wave32) per row of (16384, 4096) fp32.
// Row staged entirely in registers: 16 floats/thread. HBM traffic = 1 read + 1 write,
// both non-temporal. 12 row-wide reductions (1 max + 11 sums) each cost:
//   wave32 shuffle tree (5 ops) -> 1 LDS store -> ONE barrier -> 2x ds_load_b128
//   -> 7 VALU combine (redundant in all threads => uniform scalar everywhere).
// Double-buffered LDS partials make the single-barrier scheme hazard-free:
// reduction k writes buffer k&1; the next write to that buffer (k+2) is
// separated from this reduction's reads by reduction k+1's barrier.

#define D       4096
#define BLOCK   256
#define NWAVE   (BLOCK / 32)     // 8
#define VPT     (D / BLOCK)      // 16 floats per thread
#define N_ITER  10

typedef float v4f __attribute__((ext_vector_type(4)));

__device__ __forceinline__ float wave_sum(float v) {
#pragma unroll
    for (int off = 16; off > 0; off >>= 1)
        v += __shfl_xor(v, off, 32);
    return v;
}

__device__ __forceinline__ float wave_max(float v) {
#pragma unroll
    for (int off = 16; off > 0; off >>= 1)
        v = fmaxf(v, __shfl_xor(v, off, 32));
    return v;
}

// One barrier per reduction. `buf` is an 8-float, 16B-aligned LDS buffer,
// alternated by the caller between successive reductions.
__device__ __forceinline__ float block_sum(float v, float* buf) {
    v = wave_sum(v);
    if ((threadIdx.x & 31) == 0) buf[threadIdx.x >> 5] = v;
    __syncthreads();
    v4f a = *(const v4f*)(buf);
    v4f b = *(const v4f*)(buf + 4);
    return ((a.x + a.y) + (a.z + a.w)) + ((b.x + b.y) + (b.z + b.w));
}

__device__ __forceinline__ float block_max(float v, float* buf) {
    v = wave_max(v);
    if ((threadIdx.x & 31) == 0) buf[threadIdx.x >> 5] = v;
    __syncthreads();
    v4f a = *(const v4f*)(buf);
    v4f b = *(const v4f*)(buf + 4);
    return fmaxf(fmaxf(fmaxf(a.x, a.y), fmaxf(a.z, a.w)),
                 fmaxf(fmaxf(b.x, b.y), fmaxf(b.z, b.w)));
}

// safe_pow(z, e) = z > 0 ? z**e : 0.  SQ=true specializes e == 2 exactly
// (the alpha = 1.5 case): max(z,0)^2 -> sub/max/fma class VALU only.
template <bool SQ>
__device__ __forceinline__ float safe_pow(float z, float inv) {
    if (SQ) {
        float m = fmaxf(z, 0.0f);
        return m * m;
    }
    return (z > 0.0f) ? exp2f(inv * __log2f(z)) : 0.0f;
}

template <bool SQ>
__device__ __forceinline__ void entmax_row(const float* __restrict__ xrow,
                                           float* __restrict__ orow,
                                           float am1, float inv, float* sh) {
    // ---- stage row into registers (coalesced b128, non-temporal) ----
    const v4f* __restrict__ x4 = (const v4f*)xrow;
    v4f* __restrict__ o4 = (v4f*)orow;

    float xs[VPT];
#pragma unroll
    for (int i = 0; i < VPT / 4; ++i) {
        v4f v = __builtin_nontemporal_load(&x4[threadIdx.x + i * BLOCK]);
        xs[4 * i + 0] = v.x * am1;
        xs[4 * i + 1] = v.y * am1;
        xs[4 * i + 2] = v.z * am1;
        xs[4 * i + 3] = v.w * am1;
    }

    // ---- reduction 0: max over row ----
    float m = -INFINITY;
#pragma unroll
    for (int i = 0; i < VPT; ++i) m = fmaxf(m, xs[i]);
    m = block_max(m, sh /* buffer 0 */);

    float tau_lo = m - 1.0f;                       // 1^(alpha-1) == 1
    // pow(1/d, am1) = 2^(-12 * am1) for d = 4096
    float tau_hi = m - exp2f(am1 * __log2f(1.0f / (float)D));

    // ---- reduction 1: f_lo (computed once, never updated — faithful) ----
    float s = 0.0f;
#pragma unroll
    for (int i = 0; i < VPT; ++i) s += safe_pow<SQ>(xs[i] - tau_lo, inv);
    const float f_lo = block_sum(s, sh + 8 /* buffer 1 */) - 1.0f;

    // ---- reductions 2..11: bisection ----
    float dm    = tau_hi - tau_lo;
    float tau_m = tau_lo;
    float f_m   = 0.0f;
#pragma unroll 1
    for (int it = 0; it < N_ITER; ++it) {
        dm *= 0.5f;
        tau_m = tau_lo + dm;
        float ps = 0.0f;
#pragma unroll
        for (int i = 0; i < VPT; ++i)
            ps += safe_pow<SQ>(xs[i] - tau_m, inv);
        f_m = block_sum(ps, sh + ((it & 1) * 8)) - 1.0f;   // buffers alternate
        if (f_m * f_lo >= 0.0f) tau_lo = tau_m;            // uniform branch
    }

    // sum(p_m) is exactly the last reduction: f_m + 1
    const float rinv = 1.0f / (f_m + 1.0f);

    // ---- recompute p_m from final tau_m, normalize, stream out ----
#pragma unroll
    for (int i = 0; i < VPT / 4; ++i) {
        v4f v;
        v.x = safe_pow<SQ>(xs[4 * i + 0] - tau_m, inv) * rinv;
        v.y = safe_pow<SQ>(xs[4 * i + 1] - tau_m, inv) * rinv;
        v.z = safe_pow<SQ>(xs[4 * i + 2] - tau_m, inv) * rinv;
        v.w = safe_pow<SQ>(xs[4 * i + 3] - tau_m, inv) * rinv;
        __builtin_nontemporal_store(v, &o4[threadIdx.x + i * BLOCK]);
    }
}

__global__ __launch_bounds__(BLOCK) void entmax_bisect_kernel(
    const float* __restrict__ X, const float* __restrict__ alpha_p,
    float* __restrict__ out) {
    // Two 8-float partial buffers, 16B aligned for ds_load_b128 combines.
    __shared__ __align__(16) float sh[16];

    const size_t row = blockIdx.x;
    const float* xrow = X + row * (size_t)D;
    float*       orow = out + row * (size_t)D;

    const float am1 = alpha_p[0] - 1.0f;
    const float inv = 1.0f / am1;

    if (inv == 2.0f)                      // alpha == 1.5 fast path (uniform)
        entmax_row<true>(xrow, orow, am1, inv, sh);
    else
        entmax_row<false>(xrow, orow, am1, inv, sh);
}

extern "C" void kernel_launch(void* const* d_in, const int* in_sizes, int n_in,
                              void* d_out, int out_size, void* d_ws, size_t ws_size,
                              hipStream_t stream) {
    (void)n_in; (void)out_size; (void)d_ws; (void)ws_size;
    const float* X     = (const float*)d_in[0];
    const float* alpha = (const float*)d_in[1];
    float*       out   = (float*)d_out;

    const int rows = in_sizes[0] / D;     // 8 * 2048 = 16384
    entmax_bisect_kernel<<<dim3(rows), dim3(BLOCK), 0, stream>>>(X, alpha, out);
}